// LACLoss_45071386804580
// MI455X (gfx1250) — compile-verified
//
#include <hip/hip_runtime.h>
#include <hip/hip_bf16.h>
#include <stdint.h>

#define NCLS 20      // classes
#define KNN  16      // neighbors
#define TPB  256     // threads per block
#define ROWB (NCLS * 4)  // 80 bytes per prob row

// ---- CDNA5 async global->LDS copy (ASYNCcnt path) -------------------------
__device__ __forceinline__ void async_copy_b128_to_lds(uint32_t lds_byte_off,
                                                       const void* gptr) {
    // VDST = per-lane LDS byte offset, VADDR = per-lane 64-bit global address
    asm volatile("global_load_async_to_lds_b128 %0, %1, off"
                 :: "v"(lds_byte_off), "v"(gptr)
                 : "memory");
}

__device__ __forceinline__ void wait_async0() {
#if __has_builtin(__builtin_amdgcn_s_wait_asynccnt)
    __builtin_amdgcn_s_wait_asynccnt(0);
#else
    asm volatile("s_wait_asynccnt 0" ::: "memory");
#endif
}

// ---- Kernel 1: row softmax (N x 20), f32, memory-bound --------------------
__global__ __launch_bounds__(TPB)
void lac_softmax_kernel(const float* __restrict__ logits,
                        float* __restrict__ probs, int n) {
    int i = blockIdx.x * TPB + threadIdx.x;
    if (i >= n) return;
    const float4* lr = reinterpret_cast<const float4*>(logits + (size_t)i * NCLS);
    float x[NCLS];
#pragma unroll
    for (int c = 0; c < NCLS / 4; ++c) {
        float4 v = lr[c];
        x[4 * c + 0] = v.x; x[4 * c + 1] = v.y;
        x[4 * c + 2] = v.z; x[4 * c + 3] = v.w;
    }
    float m = x[0];
#pragma unroll
    for (int c = 1; c < NCLS; ++c) m = fmaxf(m, x[c]);
    float s = 0.0f;
#pragma unroll
    for (int c = 0; c < NCLS; ++c) { x[c] = expf(x[c] - m); s += x[c]; }
    float inv = 1.0f / s;
    float4* pr = reinterpret_cast<float4*>(probs + (size_t)i * NCLS);
#pragma unroll
    for (int c = 0; c < NCLS / 4; ++c) {
        float4 v;
        v.x = x[4 * c + 0] * inv; v.y = x[4 * c + 1] * inv;
        v.z = x[4 * c + 2] * inv; v.w = x[4 * c + 3] * inv;
        pr[c] = v;
    }
}

// ---- Kernel 2: masked pairwise prob-distance, per-block partials ----------
__global__ __launch_bounds__(TPB)
void lac_pair_kernel(const float* __restrict__ probs,
                     const int* __restrict__ labels,
                     const int* __restrict__ refidx,
                     double* __restrict__ bsum,
                     int* __restrict__ bcnt, int n) {
    __shared__ float  s_center[TPB * NCLS];   // 20 KB staged center rows
    __shared__ double s_sum[TPB];
    __shared__ int    s_cnt[TPB];

    const int tid = threadIdx.x;
    const int i   = blockIdx.x * TPB + tid;

    // Stage this thread's center prob row into LDS with async b128 copies.
    if (i < n) {
        const char* grow = reinterpret_cast<const char*>(probs + (size_t)i * NCLS);
        uint32_t lbase = (uint32_t)(uintptr_t)(const void*)(s_center + tid * NCLS);
#pragma unroll
        for (int c = 0; c < ROWB / 16; ++c)   // 5 x 16B = 80B row
            async_copy_b128_to_lds(lbase + c * 16, grow + c * 16);
    }
    wait_async0();  // each thread only reads back its own copies -> no barrier

    float  lsum = 0.0f;
    int    lcnt = 0;

    if (i < n) {
        // Center row LDS -> registers
        float p[NCLS];
        const float4* cr = reinterpret_cast<const float4*>(s_center + tid * NCLS);
#pragma unroll
        for (int c = 0; c < NCLS / 4; ++c) {
            float4 v = cr[c];
            p[4 * c + 0] = v.x; p[4 * c + 1] = v.y;
            p[4 * c + 2] = v.z; p[4 * c + 3] = v.w;
        }

        const int lab = labels[i];

        // All 16 neighbor indices in 4 coalesced int4 loads
        int idx[KNN];
        const int4* ir = reinterpret_cast<const int4*>(refidx + (size_t)i * KNN);
#pragma unroll
        for (int q = 0; q < KNN / 4; ++q) {
            int4 v = ir[q];
            idx[4 * q + 0] = v.x; idx[4 * q + 1] = v.y;
            idx[4 * q + 2] = v.z; idx[4 * q + 3] = v.w;
        }

        // 16 independent 4B label gathers (batched in flight)
        int nl[KNN];
#pragma unroll
        for (int k = 0; k < KNN; ++k) {
            int j = idx[k];
            nl[k] = labels[j < 0 ? 0 : j];
        }

        // Validity + prefetch pass: only ~1/20 of pairs survive the label
        // match, so the 80B row gather is issued only for survivors.
        unsigned vm = 0;
#pragma unroll
        for (int k = 0; k < KNN; ++k) {
            bool v = (idx[k] >= 0) & (lab != -1) & (nl[k] != -1) & (nl[k] == lab);
            if (v) {
                vm |= (1u << k);
                __builtin_prefetch(probs + (size_t)idx[k] * NCLS, 0, 3);
            }
        }

        // Compute pass
#pragma unroll
        for (int k = 0; k < KNN; ++k) {
            if ((vm >> k) & 1u) {
                const float4* qr =
                    reinterpret_cast<const float4*>(probs + (size_t)idx[k] * NCLS);
                float d = 0.0f;
#pragma unroll
                for (int c = 0; c < NCLS / 4; ++c) {
                    float4 q = qr[c];
                    float d0 = p[4 * c + 0] - q.x;
                    float d1 = p[4 * c + 1] - q.y;
                    float d2 = p[4 * c + 2] - q.z;
                    float d3 = p[4 * c + 3] - q.w;
                    d = fmaf(d0, d0, d);
                    d = fmaf(d1, d1, d);
                    d = fmaf(d2, d2, d);
                    d = fmaf(d3, d3, d);
                }
                lsum += d;
                lcnt += 1;
            }
        }
    }

    // Deterministic block tree reduction
    s_sum[tid] = (double)lsum;
    s_cnt[tid] = lcnt;
    __syncthreads();
#pragma unroll
    for (int off = TPB / 2; off > 0; off >>= 1) {
        if (tid < off) {
            s_sum[tid] += s_sum[tid + off];
            s_cnt[tid] += s_cnt[tid + off];
        }
        __syncthreads();
    }
    if (tid == 0) {
        bsum[blockIdx.x] = s_sum[0];
        bcnt[blockIdx.x] = s_cnt[0];
    }
}

// ---- Kernel 3: fixed-order final reduction -> scalar ----------------------
__global__ __launch_bounds__(TPB)
void lac_finalize_kernel(const double* __restrict__ bsum,
                         const int* __restrict__ bcnt,
                         int nb, float* __restrict__ out) {
    __shared__ double ss[TPB];
    __shared__ long long sc[TPB];
    const int tid = threadIdx.x;
    double s = 0.0;
    long long c = 0;
    for (int i = tid; i < nb; i += TPB) {  // fixed strided order
        s += bsum[i];
        c += (long long)bcnt[i];
    }
    ss[tid] = s; sc[tid] = c;
    __syncthreads();
#pragma unroll
    for (int off = TPB / 2; off > 0; off >>= 1) {
        if (tid < off) { ss[tid] += ss[tid + off]; sc[tid] += sc[tid + off]; }
        __syncthreads();
    }
    if (tid == 0) {
        double denom = (sc[0] > 0) ? (double)sc[0] : 1.0;
        out[0] = (float)(ss[0] / denom) * 1.0f;  // LOSS_WEIGHT = 1.0
    }
}

extern "C" void kernel_launch(void* const* d_in, const int* in_sizes, int n_in,
                              void* d_out, int out_size, void* d_ws, size_t ws_size,
                              hipStream_t stream) {
    const float* logits = (const float*)d_in[0];
    // d_in[1] = coord: unused (KNN indices are given)
    const int* labels = (const int*)d_in[2];
    const int* refidx = (const int*)d_in[3];
    const int n  = in_sizes[2];              // N points (labels count)
    const int nb = (n + TPB - 1) / TPB;

    // Workspace layout: [probs f32 N*20][block sums f64 nb][block counts i32 nb]
    float* probs = (float*)d_ws;
    size_t off = ((size_t)n * NCLS * sizeof(float) + 255) & ~(size_t)255;
    double* bsum = (double*)((char*)d_ws + off);
    int*    bcnt = (int*)((char*)d_ws + off + (size_t)nb * sizeof(double));

    lac_softmax_kernel<<<nb, TPB, 0, stream>>>(logits, probs, n);
    lac_pair_kernel<<<nb, TPB, 0, stream>>>(probs, labels, refidx, bsum, bcnt, n);
    lac_finalize_kernel<<<1, TPB, 0, stream>>>(bsum, bcnt, nb, (float*)d_out);
}